// SingleDimensionalHistLayer_55525337203049
// MI455X (gfx1250) — compile-verified
//
#include <hip/hip_runtime.h>

#define K_BINS 256
#define PAD 8
#define HLEN (K_BINS + 2 * PAD)   // 272 padded bins per wave histogram
#define NWAVES 8
#define TPB 256
#define EPT 8                     // elements per thread
#define EPB (TPB * EPT)           // 2048 elements per block

typedef float v2f __attribute__((ext_vector_type(2)));
typedef float v8f __attribute__((ext_vector_type(8)));

__global__ void zero_out_kernel(float* __restrict__ out, int n) {
    int i = blockIdx.x * blockDim.x + threadIdx.x;
    if (i < n) out[i] = 0.0f;
}

__global__ void __launch_bounds__(TPB)
soft_hist_kernel(const float* __restrict__ x, float* __restrict__ out,
                 int n_per_batch, float invN) {
    __shared__ float h[NWAVES * HLEN];

    const int tid  = threadIdx.x;
    const int wave = tid >> 5;
    const int lane = tid & 31;
    float* hw = &h[wave * HLEN];

    for (int i = tid; i < NWAVES * HLEN; i += TPB) h[i] = 0.0f;
    __syncthreads();

    const int  batch = blockIdx.y;
    const long base  = (long)batch * n_per_batch + (long)blockIdx.x * EPB;
    const float4* xv = (const float4*)(x + base);

    const float C   = 3.60673760222f;        // 2.5 * log2(e)
    const float K1  = 12.182493960703473f;   // e^2.5
    const float K2  = 148.4131591025766f;    // e^5
    const float K4  = 22026.465794806718f;   // e^10
    const float K8  = 485165195.40979028f;   // e^20
    const float KM1 = 11.182493960703473f;   // e^2.5 - 1

#pragma unroll
    for (int it = 0; it < EPB / 4 / TPB; ++it) {     // 2 iterations
        float4 v = xv[it * TPB + tid];
        float vals[4] = {v.x, v.y, v.z, v.w};
#pragma unroll
        for (int el = 0; el < 4; ++el) {
            float t  = vals[el] * 256.0f;            // t = x / L
            int   j0 = (int)t;                       // floor, t >= 0
            if (j0 > 255) j0 = 255;                  // safety clamp
            float f  = t - (float)j0;                // frac in [0,1)
            int idx0 = j0 + (PAD - 7);               // LDS slot of bin j0-7

            // E_i = exp(-(t-(j0-7+i))*2.5), i=0..15: one exp2 + mul ladder
            float e[16], p[16], act[16];
            e[0]  = __builtin_amdgcn_exp2f(-f * C - 7.0f * C);
            e[2]  = e[0]  * K2;
            e[4]  = e[0]  * K4;
            e[6]  = e[4]  * K2;
            e[8]  = e[0]  * K8;
            e[10] = e[8]  * K2;
            e[12] = e[8]  * K4;
            e[14] = e[12] * K2;
#pragma unroll
            for (int i = 0; i < 16; i += 2) e[i + 1] = e[i] * K1;
#pragma unroll
            for (int i = 0; i < 16; ++i) p[i] = 1.0f + e[i];

            // act_i = (K-1)*E_i / (p_i * p_{i+1}); one rcp serves two acts
#pragma unroll
            for (int gp = 0; gp < 7; ++gp) {
                int i = 2 * gp;
                float w  = __builtin_amdgcn_rcpf(p[i] * p[i + 1] * p[i + 2]);
                float wk = KM1 * w;
                act[i]     = e[i]     * (p[i + 2] * wk);
                act[i + 1] = e[i + 1] * (p[i]     * wk);
            }
            {
                float w = __builtin_amdgcn_rcpf(p[14] * p[15]);
                act[14] = e[14] * (KM1 * w);   // s_14 - s_15
                act[15] = p[14] * w;           // s_15 (bin j0+8), s_{j0+9}~2e-9
            }
#pragma unroll
            for (int i = 0; i < 16; ++i)
                atomicAdd(&hw[idx0 + i], act[i]);
        }
    }
    __syncthreads();

    // Cross-wave reduction via V_WMMA_F32_16X16X4_F32 (fp32 matrix pipe).
    // A = all-ones 16x4 -> D[m][n] = sum_k B[k][n]; B row permutation is
    // irrelevant, so 4 distinct wave histograms fill the 4 (vgpr, lane-half)
    // B slots. Two chained WMMAs sum all 8 waves in fp32.
    const int half = lane >> 4;
    const int n    = lane & 15;
    v2f a; a[0] = 1.0f; a[1] = 1.0f;

#pragma unroll
    for (int tt = 0; tt < 2; ++tt) {
        int tile = wave + tt * 8;                    // 16 tiles of 16 bins
        int boff = PAD + tile * 16 + n;

        v2f b0;
        b0[0] = h[(half * 2 + 0) * HLEN + boff];     // waves 0..3
        b0[1] = h[(half * 2 + 1) * HLEN + boff];
        v8f c = {};
        c = __builtin_amdgcn_wmma_f32_16x16x4_f32(
                false, a, false, b0, (short)0, c, false, false);

        v2f b1;
        b1[0] = h[(4 + half * 2 + 0) * HLEN + boff]; // waves 4..7
        b1[1] = h[(4 + half * 2 + 1) * HLEN + boff];
        c = __builtin_amdgcn_wmma_f32_16x16x4_f32(
                false, a, false, b1, (short)0, c, false, false);

        if (lane < 16) {
            // every D row holds the column sum; row 0 is c[0], lanes 0..15
            unsafeAtomicAdd(&out[batch * K_BINS + tile * 16 + n], c[0] * invN);
        }
    }
}

extern "C" void kernel_launch(void* const* d_in, const int* in_sizes, int n_in,
                              void* d_out, int out_size, void* d_ws, size_t ws_size,
                              hipStream_t stream) {
    const float* x   = (const float*)d_in[0];
    float*       out = (float*)d_out;

    const int total       = in_sizes[0];          // 8*512*512
    const int batches     = out_size / K_BINS;    // 8
    const int n_per_batch = total / batches;      // 262144
    const int chunks      = n_per_batch / EPB;    // 128

    zero_out_kernel<<<(out_size + 255) / 256, 256, 0, stream>>>(out, out_size);

    dim3 grid(chunks, batches);
    soft_hist_kernel<<<grid, TPB, 0, stream>>>(x, out, n_per_batch,
                                               1.0f / (float)n_per_batch);
}